// Subspace_Projection_44066364457586
// MI455X (gfx1250) — compile-verified
//
#include <hip/hip_runtime.h>
#include <math.h>

typedef __attribute__((ext_vector_type(2))) float v2f;
typedef __attribute__((ext_vector_type(4))) float v4f;
typedef __attribute__((ext_vector_type(8))) float v8f;

#define DIM_B 32
#define DIM_Q 512
#define DIM_D 640
#define DIM_C 8
#define DIM_K 5
#define N_REAL 40            // C*K
#define N_PAD 48             // padded to 3 WMMA N-tiles
#define ROWS_PER_BLOCK 64
#define THREADS 128
#define DCHUNK 64
#define N_ROWS (DIM_B * DIM_Q)   // 16384
#define W_ELEMS (DIM_D / 2 * N_PAD * 2)   // 30720 padded W floats
#define WS_G_OFF 0
#define WS_W_OFF 256             // float offset of padded W in workspace

// ---------------------------------------------------------------------------
// Kernel A: gram = W^T W via WMMA (48x48x640), discriminative loss, G_c,
// and (optionally) the padded pair-interleaved W written to workspace so the
// main kernel can stage it with coalesced float4 copies.
// One block, 256 threads = 8 waves; wave w owns d-subrange [w*16, w*16+16)
// of each 128-d chunk. Since A = W^T, the A-fragment of W^T is the same LDS
// load as the B-fragment of W -> 3 loads feed 9 WMMAs per K-step.
// ---------------------------------------------------------------------------
__global__ __launch_bounds__(256)
void gram_loss_kernel(const float* __restrict__ H,
                      float* __restrict__ Gws,      // [8*5*5]
                      float* __restrict__ Wws,      // [W_ELEMS] or nullptr
                      float* __restrict__ loss_out) // single float
{
    __shared__ __align__(16) float sW[(128 / 2) * (N_PAD * 2)]; // 6144 f
    __shared__ __align__(16) float sGram[N_PAD * N_PAD];        // 2304 f
    __shared__ float red[256];

    const int tid  = threadIdx.x;
    const int wave = tid >> 5;
    const int lane = tid & 31;
    const int grp  = lane >> 4;
    const int l15  = lane & 15;

    v8f acc[9];
#pragma unroll
    for (int t = 0; t < 9; ++t) acc[t] = (v8f){};

    for (int dc = 0; dc < DIM_D; dc += 128) {
        __syncthreads();
        // Stage 128 x 48 chunk, zero-padded, pair-interleaved layout.
        for (int idx = tid; idx < 128 * N_PAD; idx += 256) {
            int dLoc = idx / N_PAD;
            int n    = idx - dLoc * N_PAD;
            float v  = 0.f;
            if (n < N_REAL) {
                int c = n / DIM_K, k = n - c * DIM_K;
                v = H[c * (DIM_D * DIM_K) + (dc + dLoc) * DIM_K + k];
            }
            sW[(dLoc >> 1) * (N_PAD * 2) + n * 2 + (dLoc & 1)] = v;
        }
        __syncthreads();
        if (Wws) {   // contiguous chunk of global padded-W layout
            for (int idx = tid; idx < 128 * N_PAD; idx += 256)
                Wws[(dc >> 1) * (N_PAD * 2) + idx] = sW[idx];
        }
        // 4 K-steps per wave per chunk, 9 WMMAs per K-step.
#pragma unroll
        for (int dl = 0; dl < 16; dl += 4) {
            const int dk = wave * 16 + dl + grp * 2;   // even
            const float* wrow = &sW[(dk >> 1) * (N_PAD * 2)];
            v2f f[3];
            f[0] = *(const v2f*)(wrow + (0 * 16 + l15) * 2);
            f[1] = *(const v2f*)(wrow + (1 * 16 + l15) * 2);
            f[2] = *(const v2f*)(wrow + (2 * 16 + l15) * 2);
#pragma unroll
            for (int ti = 0; ti < 3; ++ti)
#pragma unroll
                for (int tj = 0; tj < 3; ++tj)
                    acc[ti * 3 + tj] = __builtin_amdgcn_wmma_f32_16x16x4_f32(
                        false, f[ti], false, f[tj], (short)0,
                        acc[ti * 3 + tj], false, false);
        }
    }

    // Deterministic cross-wave reduction of partial grams (fixed wave order).
    for (int w = 0; w < 8; ++w) {
        if (wave == w) {
#pragma unroll
            for (int t = 0; t < 9; ++t) {
                const int ti = t / 3, tj = t - ti * 3;
#pragma unroll
                for (int j = 0; j < 8; ++j) {
                    const int m = ti * 16 + grp * 8 + j;
                    const int n = tj * 16 + l15;
                    if (w == 0) sGram[m * N_PAD + n]  = acc[t][j];
                    else        sGram[m * N_PAD + n] += acc[t][j];
                }
            }
        }
        __syncthreads();
    }

    // loss = sum over off-diagonal class blocks of gram^2
    float lsum = 0.f;
    for (int e = tid; e < N_REAL * N_REAL; e += 256) {
        int n1 = e / N_REAL, n2 = e - n1 * N_REAL;
        if (n1 / DIM_K != n2 / DIM_K) {
            float gv = sGram[n1 * N_PAD + n2];
            lsum += gv * gv;
        }
    }
    red[tid] = lsum;
    __syncthreads();
    for (int s = 128; s > 0; s >>= 1) {
        if (tid < s) red[tid] += red[tid + s];
        __syncthreads();
    }
    if (tid == 0) *loss_out = red[0];

    // G_c = H_c^T H_c  (diagonal 5x5 blocks of gram)
    if (tid < DIM_C * DIM_K * DIM_K) {
        int c = tid / (DIM_K * DIM_K);
        int r = tid - c * (DIM_K * DIM_K);
        int k = r / DIM_K, m = r - k * DIM_K;
        Gws[tid] = sGram[(c * DIM_K + k) * N_PAD + (c * DIM_K + m)];
    }
}

// ---------------------------------------------------------------------------
// Main kernel: coeff GEMM via V_WMMA_F32_16X16X4_F32 + cosine epilogue.
// Block = 128 threads (4 waves), each wave owns a 16-row tile, 3 N-tiles.
// ---------------------------------------------------------------------------
__global__ __launch_bounds__(THREADS)
void subspace_sim_kernel(const float* __restrict__ X,    // [16384, 640]
                         const float* __restrict__ H,    // [8, 640, 5]
                         const float* __restrict__ Gws,  // [8, 5, 5]
                         const float* __restrict__ Wpre, // padded W or nullptr
                         float* __restrict__ out)        // [32, 8, 512]
{
    __shared__ __align__(16) float sW[(DCHUNK / 2) * (N_PAD * 2)]; // 3072 f
    __shared__ __align__(16) float sCoef[ROWS_PER_BLOCK * N_PAD];  // 3072 f
    __shared__ float sXsq[4 * 32];
    __shared__ float sG[DIM_C * DIM_K * DIM_K];

    const int tid  = threadIdx.x;
    if (tid < DIM_C * DIM_K * DIM_K) sG[tid] = Gws[tid];

    const int wave = tid >> 5;
    const int lane = tid & 31;
    const int grp  = lane >> 4;   // K-offset group: 0 -> K 0,1 ; 1 -> K 2,3
    const int l15  = lane & 15;

    const long rowBase = (long)blockIdx.x * ROWS_PER_BLOCK + wave * 16;
    const float* xrow  = X + (rowBase + l15) * DIM_D;

    v8f acc0 = {}, acc1 = {}, acc2 = {};
    float xsq = 0.f;

    for (int dc = 0; dc < DIM_D; dc += DCHUNK) {
        __syncthreads();
        if (Wpre) {  // coalesced float4 copy of precomputed padded W chunk
            const v4f* src = (const v4f*)(Wpre + (dc >> 1) * (N_PAD * 2));
            v4f* dst = (v4f*)sW;
            for (int idx = tid; idx < (DCHUNK * N_PAD) / 4; idx += THREADS)
                dst[idx] = src[idx];
        } else {     // fallback: scattered loads straight from H
            for (int idx = tid; idx < DCHUNK * N_PAD; idx += THREADS) {
                int dLoc = idx / N_PAD;
                int n    = idx - dLoc * N_PAD;
                float v  = 0.f;
                if (n < N_REAL) {
                    int c = n / DIM_K, k = n - c * DIM_K;
                    v = H[c * (DIM_D * DIM_K) + (dc + dLoc) * DIM_K + k];
                }
                sW[(dLoc >> 1) * (N_PAD * 2) + n * 2 + (dLoc & 1)] = v;
            }
        }
        __syncthreads();

        if (dc + DCHUNK < DIM_D)   // prefetch next X chunk for this row
            __builtin_prefetch(xrow + dc + DCHUNK + grp * 2, 0, 3);

#pragma unroll 4
        for (int dl = 0; dl < DCHUNK; dl += 4) {
            const int dk = dl + grp * 2;           // even
            v2f a = *(const v2f*)(xrow + dc + dk); // A frag: X[row][dk..dk+1]
            xsq += a.x * a.x + a.y * a.y;
            const float* wrow = &sW[(dk >> 1) * (N_PAD * 2)];
            v2f b0 = *(const v2f*)(wrow + (0 * 16 + l15) * 2);
            v2f b1 = *(const v2f*)(wrow + (1 * 16 + l15) * 2);
            v2f b2 = *(const v2f*)(wrow + (2 * 16 + l15) * 2);
            acc0 = __builtin_amdgcn_wmma_f32_16x16x4_f32(
                false, a, false, b0, (short)0, acc0, false, false);
            acc1 = __builtin_amdgcn_wmma_f32_16x16x4_f32(
                false, a, false, b1, (short)0, acc1, false, false);
            acc2 = __builtin_amdgcn_wmma_f32_16x16x4_f32(
                false, a, false, b2, (short)0, acc2, false, false);
        }
    }

    // C/D layout: VGPR j, lanes 0-15 -> M=j, lanes 16-31 -> M=8+j, N = lane&15
    sXsq[wave * 32 + lane] = xsq;
    const int rLoc = wave * 16 + grp * 8;
#pragma unroll
    for (int j = 0; j < 8; ++j) {
        sCoef[(rLoc + j) * N_PAD + 0 * 16 + l15] = acc0[j];
        sCoef[(rLoc + j) * N_PAD + 1 * 16 + l15] = acc1[j];
        sCoef[(rLoc + j) * N_PAD + 2 * 16 + l15] = acc2[j];
    }
    __syncthreads();

    // Epilogue: one thread per row computes 8 cosine similarities.
    if (tid < ROWS_PER_BLOCK) {
        const int r = tid;
        const long m = (long)blockIdx.x * ROWS_PER_BLOCK + r;
        const int b = (int)(m >> 9);   // / 512
        const int q = (int)(m & 511);
        const int wv = r >> 4, rl = r & 15;
        const float xn    = sXsq[wv * 32 + rl] + sXsq[wv * 32 + 16 + rl];
        const float xnorm = sqrtf(xn);
        const float* co = &sCoef[r * N_PAD];
#pragma unroll
        for (int c = 0; c < DIM_C; ++c) {
            float cv[DIM_K];
            float dot = 0.f;
#pragma unroll
            for (int k = 0; k < DIM_K; ++k) {
                cv[k] = co[c * DIM_K + k];
                dot += cv[k] * cv[k];          // x . proj == ||H^T x||^2
            }
            float pn = 0.f;                    // ||proj||^2 = c^T (H^T H) c
            const float* g = &sG[c * DIM_K * DIM_K];
#pragma unroll
            for (int k = 0; k < DIM_K; ++k)
#pragma unroll
                for (int mm = 0; mm < DIM_K; ++mm)
                    pn += g[k * DIM_K + mm] * cv[k] * cv[mm];
            float denom = fmaxf(xnorm * sqrtf(fmaxf(pn, 0.f)), 1e-8f);
            out[((long)b * DIM_C + c) * DIM_Q + q] = dot / denom;
        }
    }
}

extern "C" void kernel_launch(void* const* d_in, const int* in_sizes, int n_in,
                              void* d_out, int out_size, void* d_ws, size_t ws_size,
                              hipStream_t stream) {
    const float* X = (const float*)d_in[0];   // target_features [32,512,640]
    const float* H = (const float*)d_in[1];   // hyperplanes    [8,640,5]
    float* out = (float*)d_out;               // [32*8*512] sims + [1] loss
    float* ws  = (float*)d_ws;

    float* Gws = ws + WS_G_OFF;               // 200 floats
    const bool haveWws = ws_size >= (size_t)(WS_W_OFF + W_ELEMS) * sizeof(float);
    float* Wws = haveWws ? (ws + WS_W_OFF) : nullptr;

    gram_loss_kernel<<<1, 256, 0, stream>>>(
        H, Gws, Wws, out + (long)DIM_B * DIM_C * DIM_Q);

    dim3 grid(N_ROWS / ROWS_PER_BLOCK);       // 256 blocks
    subspace_sim_kernel<<<grid, THREADS, 0, stream>>>(
        X, H, Gws, (const float*)Wws, out);
}